// NTM_38938173505734
// MI455X (gfx1250) — compile-verified
//
#include <hip/hip_runtime.h>
#include <hip/hip_bf16.h>
#include <math.h>

// ---------------- problem constants ----------------
#define MM   524288
#define NN   64
#define NIN  256
#define HID  512
#define COUT 256
#define EPSF 1e-16f

// ---------------- workspace layout (float indices) ----------------
#define K_R     0
#define K_W     64
#define A_V     128
#define E_V     192
#define BETA_R  256
#define G_R     257
#define S_R     258   // 3 floats
#define GAMMA_R 261
#define INV_KNR 262
#define BETA_W  263
#define G_W     264
#define S_W     265   // 3 floats
#define GAMMA_W 268
#define INV_KNW 269
#define MAX_R   270   // ordered-uint
#define MAX_W   271   // ordered-uint
#define SUM_R   272
#define SUM_W   273
#define SUMW_R  274
#define SUMW_W  275
#define RDACC   276   // 64 floats
#define T_R     512
#define T_W     (512 + MM)
#define U_R     (512 + 2*MM)
#define U_W     (512 + 3*MM)

// ---------------- output layout (float indices) ----------------
#define OUT_O   0
#define MEMN_O  256
#define RDN_O   (256 + MM*64)
#define WR_O    (RDN_O + 64)
#define WW_O    (WR_O + MM)

typedef float v2f __attribute__((ext_vector_type(2)));
typedef float v8f __attribute__((ext_vector_type(8)));

__device__ __forceinline__ float sigmoidf_(float x) { return 1.0f / (1.0f + __expf(-x)); }
__device__ __forceinline__ float softplusf_(float x) { return (x > 20.0f) ? x : log1pf(__expf(x)); }

// ordered-uint encoding so unsigned atomicMax implements float max
__device__ __forceinline__ unsigned f2ord(float f) {
    unsigned u = __float_as_uint(f);
    return (u & 0x80000000u) ? ~u : (u | 0x80000000u);
}
__device__ __forceinline__ float ord2f(unsigned u) {
    return __uint_as_float((u & 0x80000000u) ? (u & 0x7fffffffu) : ~u);
}

// ============================================================
// Kernel A: controller MLP + head parameter extraction + init
// ============================================================
__global__ __launch_bounds__(256)
void ntm_controller(const float* __restrict__ x, const float* __restrict__ last_read,
                    const float* __restrict__ W1, const float* __restrict__ b1,
                    const float* __restrict__ W2, const float* __restrict__ b2,
                    const float* __restrict__ Wr, const float* __restrict__ br,
                    const float* __restrict__ Ww, const float* __restrict__ bw,
                    float* __restrict__ ws) {
    __shared__ float sIn[NIN + NN];
    __shared__ float sH[HID];
    __shared__ float sC[COUT];
    __shared__ float sPr[70];
    __shared__ float sPw[198];
    const int tid = threadIdx.x;

    for (int i = tid; i < NIN + NN; i += 256) sIn[i] = (i < NIN) ? x[i] : last_read[i - NIN];
    __syncthreads();

    for (int i = tid; i < HID; i += 256) {
        float a = b1[i];
        const float* w = W1 + (size_t)i * (NIN + NN);
        for (int j = 0; j < NIN + NN; ++j) a = fmaf(w[j], sIn[j], a);
        sH[i] = sigmoidf_(a);
    }
    __syncthreads();

    for (int i = tid; i < COUT; i += 256) {
        float a = b2[i];
        const float* w = W2 + (size_t)i * HID;
        for (int j = 0; j < HID; ++j) a = fmaf(w[j], sH[j], a);
        sC[i] = sigmoidf_(a);
    }
    __syncthreads();

    for (int i = tid; i < 70 + 198; i += 256) {
        if (i < 70) {
            float a = br[i];
            const float* w = Wr + (size_t)i * COUT;
            for (int j = 0; j < COUT; ++j) a = fmaf(w[j], sC[j], a);
            sPr[i] = a;
        } else {
            int r = i - 70;
            float a = bw[r];
            const float* w = Ww + (size_t)r * COUT;
            for (int j = 0; j < COUT; ++j) a = fmaf(w[j], sC[j], a);
            sPw[r] = a;
        }
    }
    __syncthreads();

    if (tid < 64) {
        float kr = tanhf(sPr[tid]); ws[K_R + tid] = kr; sH[tid]      = kr * kr;
        float kw = tanhf(sPw[tid]); ws[K_W + tid] = kw; sH[64 + tid] = kw * kw;
        ws[A_V + tid] = tanhf(sPw[70 + tid]);
        ws[E_V + tid] = sigmoidf_(sPw[134 + tid]);
        ws[RDACC + tid] = 0.0f;
    }
    __syncthreads();

    if (tid == 0) {
        float s1 = 0.0f, s2 = 0.0f;
        for (int j = 0; j < 64; ++j) { s1 += sH[j]; s2 += sH[64 + j]; }
        ws[INV_KNR] = 1.0f / fmaxf(sqrtf(s1), EPSF);
        ws[INV_KNW] = 1.0f / fmaxf(sqrtf(s2), EPSF);

        ws[BETA_R] = softplusf_(sPr[64]);
        ws[G_R]    = sigmoidf_(sPr[65]);
        {
            float m = fmaxf(sPr[66], fmaxf(sPr[67], sPr[68]));
            float e0 = __expf(sPr[66] - m), e1 = __expf(sPr[67] - m), e2 = __expf(sPr[68] - m);
            float inv = 1.0f / (e0 + e1 + e2);
            ws[S_R + 0] = e0 * inv; ws[S_R + 1] = e1 * inv; ws[S_R + 2] = e2 * inv;
        }
        ws[GAMMA_R] = 1.0f + softplusf_(sPr[69]);

        ws[BETA_W] = softplusf_(sPw[64]);
        ws[G_W]    = sigmoidf_(sPw[65]);
        {
            float m = fmaxf(sPw[66], fmaxf(sPw[67], sPw[68]));
            float e0 = __expf(sPw[66] - m), e1 = __expf(sPw[67] - m), e2 = __expf(sPw[68] - m);
            float inv = 1.0f / (e0 + e1 + e2);
            ws[S_W + 0] = e0 * inv; ws[S_W + 1] = e1 * inv; ws[S_W + 2] = e2 * inv;
        }
        ws[GAMMA_W] = 1.0f + softplusf_(sPw[69]);

        ((unsigned*)ws)[MAX_R] = f2ord(-INFINITY);
        ((unsigned*)ws)[MAX_W] = f2ord(-INFINITY);
        ws[SUM_R] = 0.0f; ws[SUM_W] = 0.0f;
        ws[SUMW_R] = 0.0f; ws[SUMW_W] = 0.0f;
    }
}

// ============================================================
// Kernel B: fused cosine-similarity pass for both heads (WMMA)
//   per 16-row tile: 16x v_wmma_f32_16x16x4_f32, B col0=k, col1=k2
//   B operand is tile-loop invariant -> hoisted into 32 VGPRs,
//   built branch-free with a lane mask (no in-loop LDS/EXEC churn).
//   Row norms accumulated per-lane, combined via shfl_xor(16).
// ============================================================
#define B_BLOCKS 512   // 512 blocks * 8 waves = 4096 waves; 32768 tiles -> 8 iters each

__global__ __launch_bounds__(256)
void ntm_sim(const float* __restrict__ mem, float* __restrict__ ws) {
    __shared__ float sDot[8][2][16];

    const int tid  = threadIdx.x;
    const int wv   = tid >> 5;
    const int lane = tid & 31;
    const int n    = lane & 15;      // row within tile (A) / column N (B, C/D)
    const int hi   = lane >> 4;      // K-half selector

    const float beta_r = ws[BETA_R], inv_knr = ws[INV_KNR];
    const float beta_w = ws[BETA_W], inv_knw = ws[INV_KNW];
    float* __restrict__ tR = ws + T_R;
    float* __restrict__ tW = ws + T_W;

    // ---- build loop-invariant B fragment: column 0 = k, column 1 = k2, rest 0
    const float msk = (n < 2) ? 1.0f : 0.0f;          // zero out columns >= 2
    const float* kp = ws + ((n == 1) ? K_W : K_R);    // lanes n>=2 read K_R, masked to 0
    float bx[16], by[16];
#pragma unroll
    for (int kk = 0; kk < 16; ++kk) {
        const int off = 4 * kk + 2 * hi;
        bx[kk] = msk * kp[off];
        by[kk] = msk * kp[off + 1];
    }

    const int tile0  = blockIdx.x * 8 + wv;
    const int stride = B_BLOCKS * 8;
    float lmr = -INFINITY, lmw = -INFINITY;

    for (int tile = tile0; tile < MM / 16; tile += stride) {
        const int row = tile * 16 + n;
        const float* rp = mem + (size_t)row * NN + 2 * hi;

        v8f acc = {0.f, 0.f, 0.f, 0.f, 0.f, 0.f, 0.f, 0.f};
        float sq = 0.0f;
#pragma unroll
        for (int kk = 0; kk < 16; ++kk) {
            v2f av = *(const v2f*)(rp + 4 * kk);       // global_load_b64
            sq = fmaf(av.x, av.x, fmaf(av.y, av.y, sq));
            v2f bv;
            bv.x = bx[kk];
            bv.y = by[kk];
            acc = __builtin_amdgcn_wmma_f32_16x16x4_f32(
                false, av, false, bv, (short)0, acc, false, false);
        }
        const float sqf = sq + __shfl_xor(sq, 16, 32);

        // extract D columns 0 (dot k) / 1 (dot k2) through per-wave LDS:
        // lane with (n,hi): VGPR r holds (M = 8*hi + r, N = n)
        if (n < 2) {
            float* dp = &sDot[wv][n][8 * hi];
#pragma unroll
            for (int r = 0; r < 8; ++r) dp[r] = acc[r];
        }
        __builtin_amdgcn_wave_barrier();   // intra-wave LDS is in-order; pin schedule
        if (lane < 16) {
            const float inv_mn = 1.0f / fmaxf(sqrtf(sqf), EPSF);
            const float tr = beta_r * sDot[wv][0][n] * inv_mn * inv_knr;
            const float tw = beta_w * sDot[wv][1][n] * inv_mn * inv_knw;
            tR[row] = tr;
            tW[row] = tw;
            lmr = fmaxf(lmr, tr);
            lmw = fmaxf(lmw, tw);
        }
        __builtin_amdgcn_wave_barrier();
    }

#pragma unroll
    for (int o = 16; o > 0; o >>= 1) {
        lmr = fmaxf(lmr, __shfl_xor(lmr, o, 32));
        lmw = fmaxf(lmw, __shfl_xor(lmw, o, 32));
    }
    if (lane == 0) {
        atomicMax((unsigned*)ws + MAX_R, f2ord(lmr));
        atomicMax((unsigned*)ws + MAX_W, f2ord(lmw));
    }
}

// ============================================================
// Kernel C: exp(t - max) in place, accumulate softmax denominators
// ============================================================
__global__ __launch_bounds__(256)
void ntm_exp(float* __restrict__ ws) {
    const float mr = ord2f(((const unsigned*)ws)[MAX_R]);
    const float mw = ord2f(((const unsigned*)ws)[MAX_W]);
    float* __restrict__ tR = ws + T_R;
    float* __restrict__ tW = ws + T_W;

    float sR = 0.0f, sW = 0.0f;
    for (int m = blockIdx.x * blockDim.x + threadIdx.x; m < MM; m += gridDim.x * blockDim.x) {
        float er = __expf(tR[m] - mr); tR[m] = er; sR += er;
        float ew = __expf(tW[m] - mw); tW[m] = ew; sW += ew;
    }
    __shared__ float bs[2];
    if (threadIdx.x == 0) { bs[0] = 0.0f; bs[1] = 0.0f; }
    __syncthreads();
#pragma unroll
    for (int o = 16; o > 0; o >>= 1) {
        sR += __shfl_xor(sR, o, 32);
        sW += __shfl_xor(sW, o, 32);
    }
    if ((threadIdx.x & 31) == 0) { atomicAdd(&bs[0], sR); atomicAdd(&bs[1], sW); }
    __syncthreads();
    if (threadIdx.x == 0) { atomicAdd(ws + SUM_R, bs[0]); atomicAdd(ws + SUM_W, bs[1]); }
}

// ============================================================
// Kernel D: wg = g * softmax + (1-g) * w_last   (in place)
// ============================================================
__global__ __launch_bounds__(256)
void ntm_interp(const float* __restrict__ wlr, const float* __restrict__ wlw,
                float* __restrict__ ws) {
    const float gr = ws[G_R], gw = ws[G_W];
    const float isr = 1.0f / ws[SUM_R];
    const float isw = 1.0f / ws[SUM_W];
    const int m = blockIdx.x * blockDim.x + threadIdx.x;
    if (m < MM) {
        ws[T_R + m] = gr * ws[T_R + m] * isr + (1.0f - gr) * wlr[m];
        ws[T_W + m] = gw * ws[T_W + m] * isw + (1.0f - gw) * wlw[m];
    }
}

// ============================================================
// Kernel E: circular shift + sharpen, accumulate sums of w^gamma
// ============================================================
__global__ __launch_bounds__(256)
void ntm_sharpen(float* __restrict__ ws) {
    const float s0r = ws[S_R + 0], s1r = ws[S_R + 1], s2r = ws[S_R + 2], gmr = ws[GAMMA_R];
    const float s0w = ws[S_W + 0], s1w = ws[S_W + 1], s2w = ws[S_W + 2], gmw = ws[GAMMA_W];
    const float* __restrict__ gR = ws + T_R;
    const float* __restrict__ gW = ws + T_W;
    float* __restrict__ uR = ws + U_R;
    float* __restrict__ uW = ws + U_W;

    float sR = 0.0f, sW = 0.0f;
    for (int m = blockIdx.x * blockDim.x + threadIdx.x; m < MM; m += gridDim.x * blockDim.x) {
        const int mm1 = (m == 0) ? (MM - 1) : (m - 1);
        const int mp1 = (m == MM - 1) ? 0 : (m + 1);
        float shr = gR[mm1] * s0r + gR[m] * s1r + gR[mp1] * s2r;
        float shw = gW[mm1] * s0w + gW[m] * s1w + gW[mp1] * s2w;
        float ur = __powf(shr, gmr);
        float uw = __powf(shw, gmw);
        uR[m] = ur; uW[m] = uw;
        sR += ur; sW += uw;
    }
    __shared__ float bs[2];
    if (threadIdx.x == 0) { bs[0] = 0.0f; bs[1] = 0.0f; }
    __syncthreads();
#pragma unroll
    for (int o = 16; o > 0; o >>= 1) {
        sR += __shfl_xor(sR, o, 32);
        sW += __shfl_xor(sW, o, 32);
    }
    if ((threadIdx.x & 31) == 0) { atomicAdd(&bs[0], sR); atomicAdd(&bs[1], sW); }
    __syncthreads();
    if (threadIdx.x == 0) { atomicAdd(ws + SUMW_R, bs[0]); atomicAdd(ws + SUMW_W, bs[1]); }
}

// ============================================================
// Kernel F: finalize w_r/w_w, fused read_new partials + mem_new
//   wave = 128 rows; lane l handles columns l and l+32 (coalesced 128B)
// ============================================================
#define F_BLOCKS 512
#define F_CHUNK  (MM / (F_BLOCKS * 8))   // 128 rows per wave

__global__ __launch_bounds__(256)
void ntm_update(const float* __restrict__ mem, float* __restrict__ ws,
                float* __restrict__ out) {
    float* __restrict__ memN = out + MEMN_O;
    float* __restrict__ wrO  = out + WR_O;
    float* __restrict__ wwO  = out + WW_O;

    const float iswr = 1.0f / (ws[SUMW_R] + EPSF);
    const float isww = 1.0f / (ws[SUMW_W] + EPSF);

    const int lane = threadIdx.x & 31;
    const int wv   = threadIdx.x >> 5;
    const int c0 = lane, c1 = lane + 32;
    const float a0 = ws[A_V + c0], a1 = ws[A_V + c1];
    const float e0 = ws[E_V + c0], e1 = ws[E_V + c1];

    __shared__ float cols[64];
    if (threadIdx.x < 64) cols[threadIdx.x] = 0.0f;
    __syncthreads();

    const int r0 = (blockIdx.x * 8 + wv) * F_CHUNK;
    float acc0 = 0.0f, acc1 = 0.0f;
    for (int i = 0; i < F_CHUNK; ++i) {
        const int m = r0 + i;
        const float wr  = ws[U_R + m] * iswr;
        const float wwv = ws[U_W + m] * isww;
        const size_t base = (size_t)m * NN;
        const float v0 = mem[base + c0];
        const float v1 = mem[base + c1];
        acc0 = fmaf(wr, v0, acc0);
        acc1 = fmaf(wr, v1, acc1);
        memN[base + c0] = v0 * (1.0f - wwv * e0) + wwv * a0;
        memN[base + c1] = v1 * (1.0f - wwv * e1) + wwv * a1;
        if (lane == 0) { wrO[m] = wr; wwO[m] = wwv; }
    }
    atomicAdd(&cols[c0], acc0);
    atomicAdd(&cols[c1], acc1);
    __syncthreads();
    if (threadIdx.x < 64) atomicAdd(ws + RDACC + threadIdx.x, cols[threadIdx.x]);
}

// ============================================================
// Kernel G: read_new out + final sigmoid layer
// ============================================================
__global__ __launch_bounds__(256)
void ntm_out(const float* __restrict__ x, const float* __restrict__ Wo,
             const float* __restrict__ bo, const float* __restrict__ ws,
             float* __restrict__ out) {
    __shared__ float sIn[NIN + NN];
    const int tid = threadIdx.x;
    if (tid < NIN) sIn[tid] = x[tid];
    if (tid < 64) {
        const float rn = ws[RDACC + tid];
        sIn[NIN + tid] = rn;
        out[RDN_O + tid] = rn;
    }
    __syncthreads();
    float a = bo[tid];
    const float* w = Wo + (size_t)tid * (NIN + NN);
    for (int j = 0; j < NIN + NN; ++j) a = fmaf(w[j], sIn[j], a);
    out[OUT_O + tid] = sigmoidf_(a);
}

// ============================================================
extern "C" void kernel_launch(void* const* d_in, const int* in_sizes, int n_in,
                              void* d_out, int out_size, void* d_ws, size_t ws_size,
                              hipStream_t stream) {
    (void)in_sizes; (void)n_in; (void)out_size; (void)ws_size;
    const float* x        = (const float*)d_in[0];
    const float* memory   = (const float*)d_in[1];
    const float* lastread = (const float*)d_in[2];
    const float* wlr      = (const float*)d_in[3];
    const float* wlw      = (const float*)d_in[4];
    const float* W1       = (const float*)d_in[5];
    const float* b1       = (const float*)d_in[6];
    const float* W2       = (const float*)d_in[7];
    const float* b2       = (const float*)d_in[8];
    const float* Wr       = (const float*)d_in[9];
    const float* br       = (const float*)d_in[10];
    const float* Ww       = (const float*)d_in[11];
    const float* bw       = (const float*)d_in[12];
    const float* Wo       = (const float*)d_in[13];
    const float* bo       = (const float*)d_in[14];

    float* ws  = (float*)d_ws;
    float* out = (float*)d_out;

    ntm_controller<<<1, 256, 0, stream>>>(x, lastread, W1, b1, W2, b2, Wr, br, Ww, bw, ws);
    ntm_sim<<<B_BLOCKS, 256, 0, stream>>>(memory, ws);
    ntm_exp<<<1024, 256, 0, stream>>>(ws);
    ntm_interp<<<MM / 256, 256, 0, stream>>>(wlr, wlw, ws);
    ntm_sharpen<<<1024, 256, 0, stream>>>(ws);
    ntm_update<<<F_BLOCKS, 256, 0, stream>>>(memory, ws, out);
    ntm_out<<<1, 256, 0, stream>>>(x, Wo, bo, ws, out);
}